// VoucherGraphNet_89867895701916
// MI455X (gfx1250) — compile-verified
//
#include <hip/hip_runtime.h>
#include <hip/hip_bf16.h>
#include <math.h>
#include <stdint.h>

typedef __attribute__((ext_vector_type(2))) float v2f;
typedef __attribute__((ext_vector_type(8))) float v8f;
typedef __attribute__((ext_vector_type(4))) unsigned int u32x4;
typedef __attribute__((ext_vector_type(8))) int i32x8;
typedef __attribute__((ext_vector_type(4))) int i32x4;

#define D 64
#define TPB 256
#define ROW_TILES_PER_BLOCK 8   // 8 x 16 = 128 rows per block

// ---------------- utility fills ----------------
__global__ void fill_f32_kernel(float* p, float v, int n) {
  int i = blockIdx.x * blockDim.x + threadIdx.x;
  if (i < n) p[i] = v;
}

// ---------------- embedding gather ----------------
__global__ void embed_kernel(const int* __restrict__ x, const float* __restrict__ tab,
                             const float* __restrict__ ptab, float* __restrict__ h,
                             int N, int IV, int PV) {
  int i = blockIdx.x * blockDim.x + threadIdx.x;
  if (i >= N * D) return;
  int node = i >> 6, d = i & 63;
  if (node == 0) {
    int pid = (int)(((unsigned)x[0]) % (unsigned)PV);
    h[i] = ptab[pid * D + d];
  } else {
    int vid = (int)(((unsigned)x[2 * node]) % (unsigned)IV);
    h[i] = tab[vid * D + d];
  }
}

// ---------------- edge arrays: remove self loops (mask), append self loops ----------------
__global__ void build_edges_kernel(const int* __restrict__ edge, int* __restrict__ src,
                                   int* __restrict__ dst, int* __restrict__ mask,
                                   int E, int N) {
  int i = blockIdx.x * blockDim.x + threadIdx.x;
  int tot = E + N;
  if (i >= tot) return;
  if (i < E) {
    int s = edge[i], t = edge[E + i];
    src[i] = s; dst[i] = t; mask[i] = (s != t) ? 1 : 0;
  } else {
    int l = i - E;
    src[i] = l; dst[i] = l; mask[i] = 1;
  }
}

// ---------------- scatter-add aggregation ----------------
__global__ void scatter_kernel(const float* __restrict__ h, const int* __restrict__ src,
                               const int* __restrict__ dst, const int* __restrict__ mask,
                               float* __restrict__ agg, int Ep) {
  int i = blockIdx.x * blockDim.x + threadIdx.x;
  if (i >= Ep * D) return;
  int e = i >> 6, d = i & 63;
  if (!mask[e]) return;
  atomicAdd(&agg[dst[e] * D + d], h[src[e] * D + d]);
}

// ---------------- fused WMMA GEMM: out = relu(h@Wr + agg@Wn + b) ----------------
// Block = 128 threads = 4 waves; wave w owns a 16-col slab; block covers 128 rows
// (8 row tiles). Weights are staged into LDS via the Tensor Data Mover and B
// fragments are held in registers across all row tiles.
__global__ void gemm_relu_kernel(const float* __restrict__ h, const float* __restrict__ agg,
                                 const float* __restrict__ Wr, const float* __restrict__ Wn,
                                 const float* __restrict__ bias, float* __restrict__ out, int n) {
  __shared__ float ldsW[2 * D * D];
  float* ldsWr = ldsW;
  float* ldsWn = ldsW + D * D;

#if defined(__has_builtin) && __has_builtin(__builtin_amdgcn_tensor_load_to_lds)
  if (threadIdx.x < 32) {
    // D# group0: [1:0]=count=1, [63:32]=lds_addr, [120:64]=global_addr, [127:126]=type(2)
    unsigned lr = (unsigned)(uintptr_t)ldsWr;
    unsigned ln = (unsigned)(uintptr_t)ldsWn;
    unsigned long long gr = (unsigned long long)(uintptr_t)Wr;
    unsigned long long gn = (unsigned long long)(uintptr_t)Wn;
    u32x4 g0r = {1u, lr, (unsigned)gr, (unsigned)(gr >> 32) | 0x80000000u};
    u32x4 g0n = {1u, ln, (unsigned)gn, (unsigned)(gn >> 32) | 0x80000000u};
    // D# group1: data_size=4B(code 2)@[17:16]; tensor_dim0=64@[79:48];
    // tensor_dim1=64@[111:80]; tile_dim0=64@[127:112]; tile_dim1=64@[143:128];
    // tensor_dim0_stride=64@[207:160]
    i32x8 g1 = {0x00020000, 64 << 16, 64 << 16, 64 << 16, 64, 64, 0, 0};
    i32x4 z4 = {0, 0, 0, 0};
    i32x8 z8 = {0, 0, 0, 0, 0, 0, 0, 0};
    __builtin_amdgcn_tensor_load_to_lds(g0r, g1, z4, z4, z8, 0);
    __builtin_amdgcn_tensor_load_to_lds(g0n, g1, z4, z4, z8, 0);
    __builtin_amdgcn_s_wait_tensorcnt(0);
  }
#else
  for (int i = threadIdx.x; i < D * D; i += blockDim.x) {
    ldsWr[i] = Wr[i];
    ldsWn[i] = Wn[i];
  }
#endif
  __syncthreads();

  int lane = threadIdx.x & 31;
  int wave = threadIdx.x >> 5;       // 0..3 -> col slab
  int colbase = wave * 16;
  int r = lane & 15;
  int khalf = (lane >> 4) * 2;       // 0 or 2
  int col = colbase + r;

  // Preload B fragments for the whole K range (loop-invariant across row tiles)
  v2f bR[16], bN[16];
  #pragma unroll
  for (int t = 0; t < 16; ++t) {
    int k0 = t * 4 + khalf;
    bR[t].x = ldsWr[k0 * D + col];
    bR[t].y = ldsWr[(k0 + 1) * D + col];
    bN[t].x = ldsWn[k0 * D + col];
    bN[t].y = ldsWn[(k0 + 1) * D + col];
  }
  float bc = bias[col];

  int tile0 = blockIdx.x * ROW_TILES_PER_BLOCK;
  for (int tt = 0; tt < ROW_TILES_PER_BLOCK; ++tt) {
    int rowbase = (tile0 + tt) * 16;
    if (rowbase >= n) break;                 // uniform across block
    int row = rowbase + r;
    int rowc = row < n ? row : 0;            // clamp: junk rows are never stored
    v8f acc = {};
    #pragma unroll
    for (int t = 0; t < 16; ++t) {
      int k0 = t * 4 + khalf;
      v2f a0, a1;
      a0.x = h[rowc * D + k0];   a0.y = h[rowc * D + k0 + 1];
      a1.x = agg[rowc * D + k0]; a1.y = agg[rowc * D + k0 + 1];
      acc = __builtin_amdgcn_wmma_f32_16x16x4_f32(false, a0, false, bR[t], (short)0, acc, false, false);
      acc = __builtin_amdgcn_wmma_f32_16x16x4_f32(false, a1, false, bN[t], (short)0, acc, false, false);
    }
    int rbase2 = rowbase + (lane >> 4) * 8;
    #pragma unroll
    for (int v = 0; v < 8; ++v) {
      int orow = rbase2 + v;
      if (orow < n) {
        float val = acc[v] + bc;
        out[orow * D + col] = val > 0.f ? val : 0.f;
      }
    }
  }
}

// ---------------- TopK score: s = tanh(h.p / ||p||) + order-preserving key ----------------
__global__ void score_kernel(const float* __restrict__ h, const float* __restrict__ p,
                             float* __restrict__ score, unsigned* __restrict__ key, int n) {
  int i = blockIdx.x * blockDim.x + threadIdx.x;
  if (i >= n) return;
  float dot = 0.f, nrm = 0.f;
  #pragma unroll
  for (int d = 0; d < D; ++d) {
    float pv = p[d];
    nrm += pv * pv;
    dot += h[i * D + d] * pv;
  }
  float s = tanhf(dot * rsqrtf(nrm));
  score[i] = s;
  unsigned u = __float_as_uint(s);
  u = (u & 0x80000000u) ? ~u : (u | 0x80000000u);
  key[i] = u;
}

// ---------------- 4-pass radix select of k-th largest key ----------------
// state[0]=prefix/threshold, state[1]=remaining (tie quota), state[2]=kept ctr, state[3]=tie ctr
__global__ void select_init_kernel(unsigned* state, unsigned* hist, unsigned k) {
  if (threadIdx.x == 0) { state[0] = 0; state[1] = k; state[2] = 0; state[3] = 0; }
  hist[threadIdx.x] = 0;
}

__global__ void hist_kernel(const unsigned* __restrict__ key, int n,
                            const unsigned* __restrict__ state, unsigned* __restrict__ hist,
                            int shift) {
  int i = blockIdx.x * blockDim.x + threadIdx.x;
  if (i >= n) return;
  unsigned kv = key[i];
  bool cand = (shift == 24) || ((kv >> (shift + 8)) == (state[0] >> (shift + 8)));
  if (cand) atomicAdd(&hist[(kv >> shift) & 255u], 1u);
}

__global__ void scan_kernel(unsigned* state, unsigned* hist, int shift) {
  unsigned t = state[1];
  unsigned cum = 0;
  int sel = 0;
  for (int d = 255; d >= 0; --d) {
    unsigned c = hist[d];
    if (cum + c >= t) { sel = d; break; }
    cum += c;
  }
  state[0] |= ((unsigned)sel) << shift;
  state[1] = t - cum;
  for (int i = 0; i < 256; ++i) hist[i] = 0;
}

// ---------------- compaction: keep kept nodes scaled by score, build inv ----------------
__global__ void compact_kernel(const float* __restrict__ h2, const float* __restrict__ score,
                               const unsigned* __restrict__ key, unsigned* state,
                               float* __restrict__ h, int* __restrict__ inv, int n) {
  int i = blockIdx.x * blockDim.x + threadIdx.x;
  if (i >= n) return;
  unsigned kv = key[i], thr = state[0], quota = state[1];
  bool keep = false;
  if (kv > thr) keep = true;
  else if (kv == thr) keep = (atomicAdd(&state[3], 1u) < quota);
  if (keep) {
    unsigned j = atomicAdd(&state[2], 1u);
    float s = score[i];
    #pragma unroll
    for (int d = 0; d < D; ++d) h[(int)j * D + d] = h2[i * D + d] * s;
    inv[i] = (int)j;
  } else {
    inv[i] = -1;
  }
}

// ---------------- edge relabel ----------------
__global__ void reindex_kernel(const int* __restrict__ inv, int* __restrict__ src,
                               int* __restrict__ dst, int* __restrict__ mask, int Ep) {
  int e = blockIdx.x * blockDim.x + threadIdx.x;
  if (e >= Ep) return;
  int ns = inv[src[e]], nd = inv[dst[e]];
  int m = mask[e] & (ns >= 0 ? 1 : 0) & (nd >= 0 ? 1 : 0);
  src[e] = ns < 0 ? 0 : ns;
  dst[e] = nd < 0 ? 0 : nd;
  mask[e] = m;
}

// ---------------- global max+sum pool ----------------
__device__ __forceinline__ void atomicMaxF(float* addr, float v) {
  unsigned* ua = (unsigned*)addr;
  unsigned old = *ua;
  while (true) {
    float f = __uint_as_float(old);
    if (f >= v) break;
    unsigned assumed = old;
    old = atomicCAS(ua, assumed, __float_as_uint(v));
    if (old == assumed) break;
  }
}

__global__ void pool_kernel(const float* __restrict__ h, int k,
                            float* __restrict__ pmax, float* __restrict__ psum) {
  int i = blockIdx.x * blockDim.x + threadIdx.x;
  if (i >= k * D) return;
  int d = i & 63;
  float v = h[i];
  atomicMaxF(&pmax[d], v);
  atomicAdd(&psum[d], v);
}

__global__ void accum_kernel(const float* pmax, const float* psum, float invk, float* xxg) {
  int t = threadIdx.x;
  if (t < 64) xxg[t] += pmax[t];
  else if (t < 128) xxg[t] += psum[t - 64] * invk;
}

// ---------------- final FC + sigmoid ----------------
__global__ void final_kernel(const float* __restrict__ xx, const float* __restrict__ Wl,
                             const float* __restrict__ bl, const float* __restrict__ ptab,
                             const int* __restrict__ x0, int PV, float* __restrict__ out) {
  __shared__ float red[64];
  int t = threadIdx.x;
  float acc = bl[t];
  for (int i = 0; i < 8 * D; ++i) acc += xx[i] * Wl[i * D + t];
  int pid = (int)(((unsigned)x0[0]) % (unsigned)PV);
  float emb = ptab[pid * D + t];
  out[1 + t] = emb;
  out[1 + D + t] = acc;
  red[t] = emb * acc;
  __syncthreads();
  for (int s = 32; s > 0; s >>= 1) {
    if (t < s) red[t] += red[t + s];
    __syncthreads();
  }
  if (t == 0) out[0] = 1.f / (1.f + expf(-red[0]));
}

// ===================== host side =====================
static inline int ceil_div(int a, int b) { return (a + b - 1) / b; }

extern "C" void kernel_launch(void* const* d_in, const int* in_sizes, int n_in,
                              void* d_out, int out_size, void* d_ws, size_t ws_size,
                              hipStream_t stream) {
  const int N = in_sizes[0] / 2;         // 50000
  const int E = in_sizes[1] / 2;         // 800000
  const int Ep = E + N;                  // edges + self loops
  const int IV = in_sizes[8] / D;        // item vocab
  const int PV = in_sizes[10] / D;       // promo vocab

  const float* atc_tab   = (const float*)d_in[8];
  const float* ord_tab   = (const float*)d_in[9];
  const float* promo_tab = (const float*)d_in[10];
  const float* Wl = (const float*)d_in[19];
  const float* bl = (const float*)d_in[20];

  // workspace carve-out (256B aligned)
  char* ws = (char*)d_ws;
  size_t off = 0;
  auto carve = [&](size_t bytes) -> char* {
    char* p = ws + off;
    off = (off + bytes + 255) & ~(size_t)255;
    return p;
  };
  int*      src   = (int*)carve((size_t)Ep * 4);
  int*      dst   = (int*)carve((size_t)Ep * 4);
  int*      mask  = (int*)carve((size_t)Ep * 4);
  float*    h     = (float*)carve((size_t)N * D * 4);
  float*    h2    = (float*)carve((size_t)N * D * 4);
  float*    agg   = (float*)carve((size_t)N * D * 4);
  float*    score = (float*)carve((size_t)N * 4);
  unsigned* key   = (unsigned*)carve((size_t)N * 4);
  int*      inv   = (int*)carve((size_t)N * 4);
  unsigned* hist  = (unsigned*)carve(256 * 4);
  unsigned* state = (unsigned*)carve(16 * 4);
  float*    pmax  = (float*)carve(64 * 4);
  float*    psum  = (float*)carve(64 * 4);
  float*    xx    = (float*)carve(8 * D * 4);
  (void)ws_size;

  // zero the cross-graph accumulator
  fill_f32_kernel<<<ceil_div(8 * D, TPB), TPB, 0, stream>>>(xx, 0.f, 8 * D);

  const int xi[4]  = {0, 2, 4, 6};
  const int ei[4]  = {1, 3, 5, 7};
  const float* tabs[4] = {atc_tab, ord_tab, atc_tab, ord_tab};

  for (int g = 0; g < 4; ++g) {
    const int*   xg = (const int*)d_in[xi[g]];
    const int*   eg = (const int*)d_in[ei[g]];
    const float* tb = tabs[g];

    embed_kernel<<<ceil_div(N * D, TPB), TPB, 0, stream>>>(xg, tb, promo_tab, h, N, IV, PV);
    build_edges_kernel<<<ceil_div(Ep, TPB), TPB, 0, stream>>>(eg, src, dst, mask, E, N);

    int n = N;
    for (int layer = 0; layer < 2; ++layer) {
      const float* Wr = (const float*)d_in[11 + layer * 4];
      const float* Wn = (const float*)d_in[12 + layer * 4];
      const float* bb = (const float*)d_in[13 + layer * 4];
      const float* pp = (const float*)d_in[14 + layer * 4];

      // agg = segment_sum(h[src]*mask, dst)
      fill_f32_kernel<<<ceil_div(n * D, TPB), TPB, 0, stream>>>(agg, 0.f, n * D);
      scatter_kernel<<<ceil_div(Ep * D, TPB), TPB, 0, stream>>>(h, src, dst, mask, agg, Ep);

      // h2 = relu(h@Wr + agg@Wn + b)   [WMMA f32 16x16x4 + TDM weight staging]
      gemm_relu_kernel<<<ceil_div(n, 16 * ROW_TILES_PER_BLOCK), 128, 0, stream>>>(
          h, agg, Wr, Wn, bb, h2, n);

      // scores + keys
      score_kernel<<<ceil_div(n, TPB), TPB, 0, stream>>>(h2, pp, score, key, n);

      // exact top-k threshold via 4-pass radix select
      int k = (9 * n + 9) / 10;   // ceil(0.9*n)
      select_init_kernel<<<1, 256, 0, stream>>>(state, hist, (unsigned)k);
      for (int shift = 24; shift >= 0; shift -= 8) {
        hist_kernel<<<ceil_div(n, TPB), TPB, 0, stream>>>(key, n, state, hist, shift);
        scan_kernel<<<1, 1, 0, stream>>>(state, hist, shift);
      }

      // compact kept nodes (scaled by score), build inv, relabel edges
      compact_kernel<<<ceil_div(n, TPB), TPB, 0, stream>>>(h2, score, key, state, h, inv, n);
      reindex_kernel<<<ceil_div(Ep, TPB), TPB, 0, stream>>>(inv, src, dst, mask, Ep);

      // global max + mean pool, accumulate into this graph's 128-wide slot
      fill_f32_kernel<<<1, 64, 0, stream>>>(pmax, -INFINITY, 64);
      fill_f32_kernel<<<1, 64, 0, stream>>>(psum, 0.f, 64);
      pool_kernel<<<ceil_div(k * D, TPB), TPB, 0, stream>>>(h, k, pmax, psum);
      accum_kernel<<<1, 128, 0, stream>>>(pmax, psum, 1.0f / (float)k, xx + g * 128);

      n = k;
    }
  }

  final_kernel<<<1, 64, 0, stream>>>(xx, Wl, bl, promo_tab, (const int*)d_in[0], PV,
                                     (float*)d_out);
}